// HyperbolicLogarithm_13048110645423
// MI455X (gfx1250) — compile-verified
//
#include <hip/hip_runtime.h>
#include <stdint.h>

// ---------------------------------------------------------------------------
// HyperbolicLogarithm on MI455X (gfx1250, wave32)
//
// out = A*x_proj + B*y_proj per 64-float vector. Bandwidth-bound: 768 MiB
// traffic -> ~34us floor at 23.3 TB/s. Data path: per-wave 8KB tiles staged
// with GLOBAL_LOAD_ASYNC_TO_LDS_B128 (lane-contiguous 512B/instr), padded LDS
// layout (272B stride: every b128 LDS op tiles all 64 banks in exactly 2
// passes), coefficient-form math, results returned through LDS with
// GLOBAL_STORE_ASYNC_FROM_LDS_B128 (coalesced stores).
// ---------------------------------------------------------------------------

typedef float  f32x4 __attribute__((ext_vector_type(4)));
typedef unsigned int u32x4 __attribute__((ext_vector_type(4)));
typedef int    i32x4 __attribute__((ext_vector_type(4)));
typedef int    i32x8 __attribute__((ext_vector_type(8)));

#define EPSF 1e-8f
#define VEC_D 64
#define VEC_Q 16                           // 16 x float4 per vector

#define LDS_VSTRIDE 272                    // 256B vector + 16B pad
#define WAVE_TILE_B (32 * LDS_VSTRIDE)     // one array (x or y) per wave
#define ASYNC_SMEM_BYTES (2 * 2 * WAVE_TILE_B)  // 2 waves x (x+y tiles)

// ---- kernel 0: init the zero_any flag (d_ws is poisoned, must self-init) ---
__global__ void hyp_init_flag(unsigned int* __restrict__ flag) {
    if (threadIdx.x == 0) *flag = 0u;
}

// ---- per-vector math: reductions + scalar chain ----------------------------
// On return X/Y hold the projected vectors (time slot filled) and A,B are the
// output combination coefficients: out = A*X + B*Y.
__device__ __forceinline__ void hyp_compute(
    f32x4 (&X)[VEC_Q], f32x4 (&Y)[VEC_Q],
    unsigned int* __restrict__ flag, float& A, float& B) {

    // Projection discards element 0 (replaced by time comp). Zero it so the
    // full-vector FMA reductions give exactly the spatial (d=1..63) sums.
    X[0].x = 0.0f;
    Y[0].x = 0.0f;

    float sx = 0.0f, sy = 0.0f, sxy = 0.0f;
#pragma unroll
    for (int j = 0; j < VEC_Q; ++j) {
#pragma unroll
        for (int c = 0; c < 4; ++c) {
            float xv = X[j][c], yv = Y[j][c];
            sx  = fmaf(xv, xv, sx);
            sy  = fmaf(yv, yv, sy);
            sxy = fmaf(xv, yv, sxy);
        }
    }

    float x0 = sqrtf(1.0f + sx);
    float y0 = sqrtf(1.0f + sy);

    float m   = fmaf(-x0, y0, sxy);        // mink(x, y)
    float gxx = fmaf(-x0, x0, sx);         // mink(x, x) ~ -1
    float gyy = fmaf(-y0, y0, sy);         // mink(y, y) ~ -1

    float inner = fmaxf(-m, 1.0f + EPSF);
    float dist  = acoshf(inner);
    if (dist < EPSF) atomicOr(flag, 1u);   // feeds global zero_any

    // v = a*x + b*y coefficient tracking through the whole chain.
    float a = inner, b = 1.0f;
    float q = fmaf(a * a, gxx, fmaf(2.0f * a * b, m, b * b * gyy));
    float s = dist / sqrtf(fmaxf(q, EPSF));
    a *= s; b *= s;

    float in1 = fmaf(a, gxx, b * m);
    if (fabsf(in1) > EPSF) a += in1 * (1.0f / (1.0f + EPSF));
    float chk = fmaf(a, gxx, b * m);
    if (fabsf(chk) > EPSF) a -= chk * (1.0f / (1.0f + EPSF));

    float vsq = fmaf(a * a, gxx, fmaf(2.0f * a * b, m, b * b * gyy));
    float t = dist / sqrtf(fmaxf(-vsq, EPSF));
    A = t * a;
    B = t * b;

    X[0].x = x0;                           // projected time components
    Y[0].x = y0;
}

__device__ __forceinline__ void hyp_direct_one(
    const float* __restrict__ xg, const float* __restrict__ yg,
    float* __restrict__ outg, unsigned int* __restrict__ flag, long v) {
    const f32x4* xp = (const f32x4*)xg + (size_t)v * VEC_Q;
    const f32x4* yp = (const f32x4*)yg + (size_t)v * VEC_Q;
    f32x4 X[VEC_Q], Y[VEC_Q];
#pragma unroll
    for (int j = 0; j < VEC_Q; ++j) X[j] = __builtin_nontemporal_load(xp + j);
#pragma unroll
    for (int j = 0; j < VEC_Q; ++j) Y[j] = __builtin_nontemporal_load(yp + j);
    float A, B;
    hyp_compute(X, Y, flag, A, B);
    f32x4* op = (f32x4*)outg + (size_t)v * VEC_Q;
#pragma unroll
    for (int j = 0; j < VEC_Q; ++j) {
        f32x4 r = A * X[j] + B * Y[j];
        __builtin_nontemporal_store(r, op + j);
    }
}

// ---- kernel 1: async-LDS staged main kernel (CDNA5 async data path) --------
// Host pass sees an empty stub (no device asm sema on x86); device pass gets
// the real body. Launched unconditionally.
__global__ __launch_bounds__(64) void hyp_log_main_async(
    const float* __restrict__ xg, const float* __restrict__ yg,
    float* __restrict__ outg, unsigned int* __restrict__ flag, int nvec) {
#if defined(__HIP_DEVICE_COMPILE__)
    extern __shared__ __attribute__((aligned(16))) char smem[];
    const int  lane     = threadIdx.x & 31;
    const int  wave     = threadIdx.x >> 5;
    const long waveVec0 = (long)blockIdx.x * 64 + (long)wave * 32;

    if (waveVec0 + 32 <= (long)nvec) {
        char* ldsx = smem + wave * (2 * WAVE_TILE_B);
        char* ldsy = ldsx + WAVE_TILE_B;
        // Low 32 bits of a generic LDS pointer are the LDS byte offset.
        const unsigned ldsx_off = (unsigned)(uintptr_t)ldsx;
        const unsigned ldsy_off = (unsigned)(uintptr_t)ldsy;
        const unsigned long long gx =
            (unsigned long long)(uintptr_t)(xg + waveVec0 * VEC_D);
        const unsigned long long gy =
            (unsigned long long)(uintptr_t)(yg + waveVec0 * VEC_D);

        // Stage 8KB x-tile + 8KB y-tile: lane-contiguous 512B per instruction,
        // LDS dst scattered into the padded per-vector layout.
#pragma unroll
        for (int k = 0; k < 16; ++k) {
            int chunk = k * 32 + lane;                       // 16B chunk id
            unsigned voff =
                (unsigned)((chunk >> 4) * LDS_VSTRIDE + (chunk & 15) * 16);
            unsigned long long goff = (unsigned long long)(chunk * 16);
            asm volatile("global_load_async_to_lds_b128 %0, %1, off"
                         :: "v"(ldsx_off + voff), "v"(gx + goff) : "memory");
            asm volatile("global_load_async_to_lds_b128 %0, %1, off"
                         :: "v"(ldsy_off + voff), "v"(gy + goff) : "memory");
        }
        asm volatile("s_wait_asynccnt 0" ::: "memory");

        // Each lane pulls its own vector (272B stride -> 2-pass b128, optimal).
        const char* mx = ldsx + lane * LDS_VSTRIDE;
        const char* my = ldsy + lane * LDS_VSTRIDE;
        f32x4 X[VEC_Q], Y[VEC_Q];
#pragma unroll
        for (int j = 0; j < VEC_Q; ++j) {
            X[j] = *(const f32x4*)(mx + j * 16);
            Y[j] = *(const f32x4*)(my + j * 16);
        }

        float A, B;
        hyp_compute(X, Y, flag, A, B);

        // Results into the (now dead) x tile, then coalesced async stores.
        char* rx = ldsx + lane * LDS_VSTRIDE;
#pragma unroll
        for (int j = 0; j < VEC_Q; ++j) {
            f32x4 r = A * X[j] + B * Y[j];
            *(f32x4*)(rx + j * 16) = r;
        }
        asm volatile("s_wait_dscnt 0" ::: "memory");  // LDS data visible to DMA

        const unsigned long long go =
            (unsigned long long)(uintptr_t)(outg + waveVec0 * VEC_D);
#pragma unroll
        for (int k = 0; k < 16; ++k) {
            int chunk = k * 32 + lane;
            unsigned voff =
                (unsigned)((chunk >> 4) * LDS_VSTRIDE + (chunk & 15) * 16);
            asm volatile("global_store_async_from_lds_b128 %0, %1, off"
                         :: "v"(go + (unsigned long long)(chunk * 16)),
                            "v"(ldsx_off + voff) : "memory");
        }
        // s_endpgm performs an implicit wait-idle, covering ASYNCcnt.
    } else {
        // Tail fallback (unused for the harness shape): direct loads/stores.
        long v = waveVec0 + lane;
        if (v < (long)nvec) hyp_direct_one(xg, yg, outg, flag, v);
    }
#endif  // __HIP_DEVICE_COMPILE__
}

// ---- kernel 1b: direct-load main kernel (kept as reference/fallback) -------
__global__ __launch_bounds__(256) void hyp_log_main_direct(
    const float* __restrict__ xg, const float* __restrict__ yg,
    float* __restrict__ outg, unsigned int* __restrict__ flag, int nvec) {
    int v = blockIdx.x * 256 + threadIdx.x;
    if (v >= nvec) return;
    hyp_direct_one(xg, yg, outg, flag, v);
}

// ---- kernel 2: zero entire output iff any dist < EPS -----------------------
__global__ __launch_bounds__(256) void hyp_zero_fix(
    float* __restrict__ outg, const unsigned int* __restrict__ flag, long n4) {
    if (*flag == 0u) return;
    f32x4* o = (f32x4*)outg;
    long i      = (long)blockIdx.x * 256 + threadIdx.x;
    long stride = (long)gridDim.x * 256;
    f32x4 z = {0.0f, 0.0f, 0.0f, 0.0f};
    for (; i < n4; i += stride) __builtin_nontemporal_store(z, o + i);
}

// ---------------------------------------------------------------------------
// TDM probe (NOT launched): keeps tensor_load_to_lds / s_wait_tensorcnt
// lowering verified on this toolchain.
// ---------------------------------------------------------------------------
#if defined(__has_builtin)
#if __has_builtin(__builtin_amdgcn_tensor_load_to_lds)
__global__ void hyp_tdm_probe() {
    u32x4 g0 = {0u, 0u, 0u, 0u};
    i32x8 g1 = {0, 0, 0, 0, 0, 0, 0, 0};
    i32x4 g2 = {0, 0, 0, 0};
    i32x4 g3 = {0, 0, 0, 0};
#if __has_include(<hip/amd_detail/amd_gfx1250_TDM.h>)
    i32x8 g4 = {0, 0, 0, 0, 0, 0, 0, 0};
    __builtin_amdgcn_tensor_load_to_lds(g0, g1, g2, g3, g4, 0);
#else
    __builtin_amdgcn_tensor_load_to_lds(g0, g1, g2, g3, 0);
#endif
#if __has_builtin(__builtin_amdgcn_s_wait_tensorcnt)
    __builtin_amdgcn_s_wait_tensorcnt(0);
#endif
}
#endif
#endif  // __has_builtin

// ---------------------------------------------------------------------------
extern "C" void kernel_launch(void* const* d_in, const int* in_sizes, int n_in,
                              void* d_out, int out_size, void* d_ws, size_t ws_size,
                              hipStream_t stream) {
    const float* x = (const float*)d_in[0];
    const float* y = (const float*)d_in[1];
    float* out = (float*)d_out;
    unsigned int* flag = (unsigned int*)d_ws;

    int nvec = in_sizes[0] / VEC_D;          // 2048*512 = 1,048,576

    hyp_init_flag<<<1, 32, 0, stream>>>(flag);

    int blocks = (nvec + 63) / 64;           // 64 vectors (2 waves) per block
    hyp_log_main_async<<<blocks, 64, ASYNC_SMEM_BYTES, stream>>>(
        x, y, out, flag, nvec);

    long n4 = (long)out_size / 4;
    hyp_zero_fix<<<2048, 256, 0, stream>>>(out, flag, n4);
}